// WalkLayer_37366215475447
// MI455X (gfx1250) — compile-verified
//
#include <hip/hip_runtime.h>
#include <hip/hip_bf16.h>
#include <stddef.h>

// WalkLayer forward for B=2, N=64, K=64, beta=0.9, relu activation.
//
// Stage 1: g1 = graph @ W  -> FP32 WMMA (V_WMMA_F32_16X16X4_F32), wave32.
// Stage 2: out[b,i,m,k] = lerp over j-reduction of relu(g1[b,i,j,k]*graph[b,j,m,k])
//          with diagonal + bool-mask zeroing. relu-inside-sum => VALU, not GEMM.
// Whole working set (~8MB) is L2-resident on MI455X (192MB L2); problem is
// latency-bound, so we fuse everything into two kernels and never build g5.

typedef __attribute__((ext_vector_type(2))) float v2f;
typedef __attribute__((ext_vector_type(8))) float v8f;

#define WL_B 2
#define WL_N 64
#define WL_K 64
#define WL_BETA 0.9f

// ---------------------------------------------------------------------------
// Kernel 1: g1[r, l] = sum_k graph[r, k] * W[k, l], r in [0, B*N*N)
// One wave32 per 16x16 output tile; K reduced in 16 steps of WMMA 16x16x4 f32.
// ---------------------------------------------------------------------------
__global__ void __launch_bounds__(128)
walk_g1_wmma(const float* __restrict__ graph,   // [8192, 64] row-major (A)
             const float* __restrict__ W,       // [64, 64]   row-major (B)
             float* __restrict__ g1)            // [8192, 64] row-major (D)
{
    __shared__ float sW[WL_K * WL_K];           // 16 KB of the 320 KB WGP LDS

    const int tid = threadIdx.x;
    // Stage W into LDS: 4096 floats / 128 threads = 8 float4 per thread.
    {
        const float4* Wv  = (const float4*)W;
        float4*       sWv = (float4*)sW;
        #pragma unroll
        for (int idx = tid; idx < (WL_K * WL_K) / 4; idx += 128)
            sWv[idx] = Wv[idx];
    }
    __syncthreads();

    const int wave = tid >> 5;                  // 0..3 waves per block
    const int lane = tid & 31;
    const int half = lane >> 4;                 // 0: lanes 0-15, 1: lanes 16-31
    const int lr   = lane & 15;

    const int tile    = blockIdx.x * 4 + wave;  // 2048 tiles total
    const int rowTile = tile >> 2;              // 512 row tiles (16 rows each)
    const int colTile = tile & 3;               // 4 col tiles  (16 cols each)
    const int r0 = rowTile * 16;
    const int c0 = colTile * 16;

    v8f acc = {};                               // 16x16 f32 accumulator (8 VGPRs)
    const float* arow = graph + (size_t)(r0 + lr) * WL_K;

    #pragma unroll
    for (int ks = 0; ks < WL_K / 4; ++ks) {
        const int kb = ks * 4;
        // A frag (ISA 32-bit A 16x4): lane lr holds row M=lr;
        // VGPR v holds K = kb + v + 2*half  -> contiguous pair, b64 load.
        v2f a = *(const v2f*)(arow + kb + 2 * half);
        // B frag (4x16, row striped across lanes): VGPR v = B[kb+v+2*half, c0+lr]
        v2f b;
        b.x = sW[(kb + 2 * half + 0) * WL_K + c0 + lr];
        b.y = sW[(kb + 2 * half + 1) * WL_K + c0 + lr];
        // 8-arg form: (neg_a, A, neg_b, B, c_mod, C, reuse_a, reuse_b)
        acc = __builtin_amdgcn_wmma_f32_16x16x4_f32(
            false, a, false, b, (short)0, acc, false, false);
    }

    // D layout (16x16 f32 C/D): VGPR v -> row M = v + 8*half, col N = c0 + lr.
    #pragma unroll
    for (int v = 0; v < 8; ++v)
        g1[(size_t)(r0 + v + 8 * half) * WL_K + c0 + lr] = acc[v];
}

// ---------------------------------------------------------------------------
// Kernel 2: per (b,i,m) block, 64 threads (one k each):
//   new = sum_j relu(g1[b,i,j,k] * graph[b,j,m,k])  skipping j==i, j==m,
//         !mask_[b,i,j,m];  i==m zeroes everything -> out = graph.
//   beta = (all j-terms exactly 0) ? 1.0 : 0.9
//   out  = new + beta * (graph - new)
// ---------------------------------------------------------------------------
__global__ void __launch_bounds__(64)
walk_aggregate(const float* __restrict__ graph,           // [B,N,N,K]
               const float* __restrict__ g1,              // [B,N,N,K]
               const unsigned char* __restrict__ mask,    // [B,N,N,N] (numpy bool)
               float* __restrict__ out)                   // [B,N,N,K]
{
    const int k = threadIdx.x;                 // 0..63
    const int m = blockIdx.x & (WL_N - 1);
    const int i = (blockIdx.x >> 6) & (WL_N - 1);
    const int b = blockIdx.x >> 12;

    const size_t outIdx = (((size_t)(b * WL_N + i)) * WL_N + m) * WL_K + k;
    const float gval = graph[outIdx];

    if (i == m) {               // i==m diagonal zeroes g5 for all j,k => beta=1
        out[outIdx] = gval;
        return;
    }

    const float* g1row = g1    + ((size_t)(b * WL_N + i)) * WL_N * WL_K + k;   // +j*64
    const float* grcol = graph + ((size_t)b) * WL_N * WL_N * WL_K + m * WL_K + k; // +j*4096
    const unsigned char* mrow =
        mask + ((size_t)(b * WL_N + i)) * WL_N * WL_N + m;                     // +j*64

    float acc = 0.0f;
    bool  nz  = false;
    #pragma unroll 4
    for (int j = 0; j < WL_N; ++j) {
        if (j == i || j == m) continue;        // j==i / j==m diagonals
        if (!mrow[(size_t)j * WL_N]) continue; // mask_[b,i,j,m]
        const float val = g1row[(size_t)j * WL_K] * grcol[(size_t)j * (WL_N * WL_K)];
        nz  |= (val != 0.0f);                  // beta condition: ALL terms == 0
        acc += fmaxf(val, 0.0f);               // relu then sum over j
    }

    const float beta = nz ? WL_BETA : 1.0f;
    out[outIdx] = acc + beta * (gval - acc);   // lerp(new, old, beta)
}

// ---------------------------------------------------------------------------
extern "C" void kernel_launch(void* const* d_in, const int* in_sizes, int n_in,
                              void* d_out, int out_size, void* d_ws, size_t ws_size,
                              hipStream_t stream) {
    const float*         graph = (const float*)d_in[0];          // [2,64,64,64] f32
    const float*         W     = (const float*)d_in[1];          // [64,64]      f32
    const unsigned char* mask  = (const unsigned char*)d_in[2];  // [2,64,64,64] bool (1B)
    float*               outp  = (float*)d_out;                  // [2,64,64,64] f32
    float*               g1    = (float*)d_ws;                   // 2 MB scratch

    // 8192x64 GEMM: 2048 16x16 tiles, 4 waves (128 threads) per block.
    walk_g1_wmma<<<512, 128, 0, stream>>>(graph, W, g1);
    // 8192 (b,i,m) blocks, 64 threads (k) each.
    walk_aggregate<<<WL_B * WL_N * WL_N, WL_K, 0, stream>>>(graph, g1, mask, outp);
}